// GCNAggregator_60653528154230
// MI455X (gfx1250) — compile-verified
//
#include <hip/hip_runtime.h>

typedef float v2f __attribute__((ext_vector_type(2)));
typedef float v4f __attribute__((ext_vector_type(4)));
typedef float v8f __attribute__((ext_vector_type(8)));

#define D_FEAT 128
#define N_SAMP 25
#define ROWS_PER_BLOCK 32
#define A_STRIDE 132  // 128 + 4 pad: keeps ds_load_b64 A-frag reads bank-conflict free

// Fused: neigh = mean_s(features[idx[b,t,s]]) ; out = relu(neigh @ W)
// Block = 256 threads (8 waves), owns 32 output rows (two 16x16 WMMA M-tiles).
// Wave w owns N-columns [16w, 16w+16), K=128 swept in 32 steps of V_WMMA_F32_16X16X4_F32.
__global__ __launch_bounds__(256)
void gcn_agg_wmma_f32(const float* __restrict__ features,
                      const long long* __restrict__ sample,
                      const float* __restrict__ W,
                      float* __restrict__ out)
{
    __shared__ float Atile[ROWS_PER_BLOCK * A_STRIDE];

    const int tid  = threadIdx.x;
    const int lane = tid & 31;
    const int wv   = tid >> 5;        // wave id 0..7 -> N-tile
    const int half = lane >> 4;       // K-half selector (ISA 7.12.2 layout)
    const int sidx = lane & 15;       // M (for A) or N (for B/D) index in tile
    const int ncol = wv * 16 + sidx;  // global N column for this lane

    // ---------------- preload B fragments (weights) into registers ----------------
    // b[v] for k-step kb holds W[k][ncol], k = 4*kb + 2*half + v  (B: 4x16, lane=N,
    // K-halves split across lane groups, VGPR index walks K within the half).
    float bfrag[64];
#pragma unroll
    for (int kb = 0; kb < 32; ++kb) {
#pragma unroll
        for (int v = 0; v < 2; ++v) {
            const int k = kb * 4 + half * 2 + v;
            bfrag[kb * 2 + v] = W[k * D_FEAT + ncol];
        }
    }

    // ---------------- phase 1: gather + mean -> LDS A tile ----------------
    // thread -> (row_in_tile, 16-col strip); 8 threads cover one 512B feature row.
    const int rit = tid >> 3;                 // 0..31
    const int cb  = (tid & 7) << 4;           // 0,16,...,112
    const long long grow = (long long)blockIdx.x * ROWS_PER_BLOCK + rit;
    const long long* idxp = sample + grow * N_SAMP;

    v4f acc0 = {0.f, 0.f, 0.f, 0.f};
    v4f acc1 = {0.f, 0.f, 0.f, 0.f};
    v4f acc2 = {0.f, 0.f, 0.f, 0.f};
    v4f acc3 = {0.f, 0.f, 0.f, 0.f};
    for (int s = 0; s < N_SAMP; ++s) {
        const long long nid = idxp[s];
        const v4f* fp = (const v4f*)(features + nid * (long long)D_FEAT + cb);
        acc0 += fp[0];
        acc1 += fp[1];
        acc2 += fp[2];
        acc3 += fp[3];
    }
    const float sc = 1.0f / (float)N_SAMP;
    acc0 *= sc; acc1 *= sc; acc2 *= sc; acc3 *= sc;

    v4f* dst = (v4f*)&Atile[rit * A_STRIDE + cb];  // 16B aligned (528*rit + 64*(tid&7))
    dst[0] = acc0; dst[1] = acc1; dst[2] = acc2; dst[3] = acc3;

    __syncthreads();

    // ---------------- phase 2: K-loop of f32 WMMA ----------------
    // A: 16x4 f32, lane 0-15 -> M=lane holds K={k0,k0+1}; lanes 16-31 -> K+=2.
    const float* arow0 = &Atile[sidx * A_STRIDE];         // M-tile 0 (rows 0..15)
    const float* arow1 = &Atile[(sidx + 16) * A_STRIDE];  // M-tile 1 (rows 16..31)

    v8f c0 = {0.f, 0.f, 0.f, 0.f, 0.f, 0.f, 0.f, 0.f};
    v8f c1 = {0.f, 0.f, 0.f, 0.f, 0.f, 0.f, 0.f, 0.f};

#pragma unroll
    for (int kb = 0; kb < 32; ++kb) {
        const int k0 = kb * 4 + half * 2;                 // even -> 8B-aligned ds_load_b64
        const v2f aA = *(const v2f*)(arow0 + k0);
        const v2f aB = *(const v2f*)(arow1 + k0);
        v2f b;
        b.x = bfrag[2 * kb];
        b.y = bfrag[2 * kb + 1];
        // (neg_a, A, neg_b, B, c_mod, C, reuse_a, reuse_b)
        c0 = __builtin_amdgcn_wmma_f32_16x16x4_f32(false, aA, false, b, (short)0, c0,
                                                   false, false);
        c1 = __builtin_amdgcn_wmma_f32_16x16x4_f32(false, aB, false, b, (short)0, c1,
                                                   false, false);
    }

    // ---------------- ReLU + store ----------------
    // D layout: VGPR r -> M = r + 8*half (c0: rows 0..15, c1: rows 16..31), N = ncol.
    float* op = out + (size_t)blockIdx.x * ROWS_PER_BLOCK * D_FEAT;
#pragma unroll
    for (int r = 0; r < 8; ++r) {
        const int m = r + half * 8;
        float x = c0[r];
        op[m * D_FEAT + ncol] = x > 0.f ? x : 0.f;
        float y = c1[r];
        op[(m + 16) * D_FEAT + ncol] = y > 0.f ? y : 0.f;
    }
}

extern "C" void kernel_launch(void* const* d_in, const int* in_sizes, int n_in,
                              void* d_out, int out_size, void* d_ws, size_t ws_size,
                              hipStream_t stream) {
    (void)n_in; (void)d_ws; (void)ws_size; (void)out_size;
    const float*     features = (const float*)d_in[0];      // [100000,128] f32
    const long long* sample   = (const long long*)d_in[1];  // [8,4096,25] i64
    const float*     W        = (const float*)d_in[2];      // [128,128] f32
    float*           out      = (float*)d_out;              // [8,4096,128] f32

    const int total_rows = in_sizes[1] / N_SAMP;            // B*T = 32768
    const int blocks = total_rows / ROWS_PER_BLOCK;         // 1024
    gcn_agg_wmma_f32<<<dim3(blocks), dim3(256), 0, stream>>>(features, sample, W, out);
}